// GNN_44178033607203
// MI455X (gfx1250) — compile-verified
//
#include <hip/hip_runtime.h>
#include <math.h>

typedef __attribute__((ext_vector_type(2))) float v2f;
typedef __attribute__((ext_vector_type(4))) float v4f;
typedef __attribute__((ext_vector_type(8))) float v8f;

#define FDIM 128
#define PITCH 132  // 128 + 4 pad: bank stride 4 -> conflict-free b64 fragment reads

// ---------------- degree / normalization ----------------
__global__ void k_deg_init(float* deg, int N) {
  int n = blockIdx.x * blockDim.x + threadIdx.x;
  if (n < N) deg[n] = 1.0f;  // self-loop
}

__global__ void k_deg_count(const int* __restrict__ dst, float* deg, int E) {
  int e = blockIdx.x * blockDim.x + threadIdx.x;
  if (e < E) atomicAdd(&deg[dst[e]], 1.0f);
}

__global__ void k_dinv(float* deg, int N) {
  int n = blockIdx.x * blockDim.x + threadIdx.x;
  if (n < N) deg[n] = rsqrtf(deg[n]);  // deg >= 1 always (self-loops)
}

// ---------------- WMMA f32 GEMM: H[N,128] = X[N,128] @ W[128,128] ----------------
// 256 threads = 8 waves. X tile (128 rows) staged via async global->LDS b128
// copies (ASYNCcnt path); W staged transposed via regular DS stores while the
// async copies are in flight. Wave w owns rows [w*16,w*16+16); 8 col tiles x
// 32 K-steps of v_wmma_f32_16x16x4_f32, B fragment = one ds_load_b64.
__global__ __launch_bounds__(256) void k_gemm128(const float* __restrict__ X,
                                                 const float* __restrict__ W,
                                                 float* __restrict__ Hout,
                                                 int N) {
  __shared__ float sA[128 * PITCH];   // X tile, row-major
  __shared__ float sWt[128 * PITCH];  // W transposed: sWt[col*PITCH + k]
  const int tid = threadIdx.x;
  const int rowbase = blockIdx.x * 128;

  // Async-stage X tile: 4096 16B chunks, 16 per thread. LDS row offset
  // row*PITCH*4 = row*528 B (16B aligned); rows are 512B in global (aligned).
  for (int j = tid; j < 128 * 32; j += 256) {
    int row = j >> 5;
    int c = (j & 31) * 4;
    int gr = rowbase + row;
    if (gr >= N) gr = N - 1;  // clamp: padded rows masked at store
    unsigned lds_dst = (unsigned)(unsigned long long)&sA[row * PITCH + c];
    unsigned long long ga = (unsigned long long)(X + (size_t)gr * FDIM + c);
    asm volatile("global_load_async_to_lds_b128 %0, %1, off"
                 :: "v"(lds_dst), "v"(ga) : "memory");
  }

  // Transpose-stage W (overlaps the async copies): b128 global reads,
  // scatter to sWt[col][k].
  for (int i = tid; i < (128 * 128) / 4; i += 256) {
    int r = (4 * i) >> 7;
    int c0 = (4 * i) & 127;
    v4f w4 = ((const v4f*)W)[i];
    sWt[(c0 + 0) * PITCH + r] = w4.x;
    sWt[(c0 + 1) * PITCH + r] = w4.y;
    sWt[(c0 + 2) * PITCH + r] = w4.z;
    sWt[(c0 + 3) * PITCH + r] = w4.w;
  }

  asm volatile("s_wait_asynccnt 0x0" ::: "memory");
  __syncthreads();

  const int wave = tid >> 5;
  const int lane = tid & 31;
  const int m = lane & 15;   // M (for A) / N (for B,C,D) within tile
  const int kh = lane >> 4;  // K-half select per ISA layout
  const float* aRow = &sA[(wave * 16 + m) * PITCH];

  for (int nt = 0; nt < 8; ++nt) {
    const int col = nt * 16 + m;
    const float* bCol = &sWt[col * PITCH];
    v8f acc = {};
#pragma unroll
    for (int kk = 0; kk < 32; ++kk) {
      const int k0 = 4 * kk + 2 * kh;
      v2f a = *(const v2f*)&aRow[k0];  // A: 16x4 lane-striped (hoisted to regs)
      v2f b = *(const v2f*)&bCol[k0];  // B: 4x16, contiguous in sWt -> ds_load_b64
      acc = __builtin_amdgcn_wmma_f32_16x16x4_f32(
          /*neg_a=*/false, a, /*neg_b=*/false, b,
          /*c_mod=*/(short)0, acc, /*reuse_a=*/false, /*reuse_b=*/false);
    }
#pragma unroll
    for (int r = 0; r < 8; ++r) {  // C/D: VGPR r -> M = r + 8*kh
      int row = rowbase + wave * 16 + r + 8 * kh;
      if (row < N) Hout[(size_t)row * FDIM + col] = acc[r];
    }
  }
}

// ---------------- scatter phase ----------------
// Init out with the self-loop term (also deterministically overwrites stale
// workspace each call): out[n] = h[n] * dinv[n]^2. float4 per thread.
__global__ void k_selfloop(const float* __restrict__ h, const float* __restrict__ dinv,
                           float* __restrict__ out, int N) {
  size_t i = (size_t)blockIdx.x * blockDim.x + threadIdx.x;  // float4 index
  if (i < (size_t)N * 32) {
    int n = (int)(i >> 5);
    float di = dinv[n];
    v4f h4 = ((const v4f*)h)[i];
    ((v4f*)out)[i] = h4 * (di * di);
  }
}

// One wave per edge: 32 lanes x float4 = 128 features. src/dst/dinv loads are
// wave-uniform (scalarized); atomics are address-coalesced f32 adds.
__global__ void k_edge_scatter(const float* __restrict__ h, const int* __restrict__ src,
                               const int* __restrict__ dst, const float* __restrict__ dinv,
                               float* out, long long E) {
  long long i = (long long)blockIdx.x * blockDim.x + threadIdx.x;
  if (i < E * 32) {
    int e = (int)(i >> 5);
    int fq = (int)(i & 31);
    int s = src[e], d = dst[e];
    float w = dinv[s] * dinv[d];
    v4f h4 = ((const v4f*)h)[(size_t)s * 32 + fq];
    float* o = &out[(size_t)d * FDIM + 4 * fq];
    atomicAdd(o + 0, h4.x * w);
    atomicAdd(o + 1, h4.y * w);
    atomicAdd(o + 2, h4.z * w);
    atomicAdd(o + 3, h4.w * w);
  }
}

__global__ void k_bias_relu(const float* __restrict__ in, const float* __restrict__ b,
                            float* __restrict__ out, int N) {
  size_t i = (size_t)blockIdx.x * blockDim.x + threadIdx.x;  // float4 index
  if (i < (size_t)N * 32) {
    int fq = (int)(i & 31);
    v4f v = ((const v4f*)in)[i] + ((const v4f*)b)[fq];
    v4f z = {};
    ((v4f*)out)[i] = v > z ? v : z;  // elementwise select on ext vectors
  }
}

// ---------------- pooling + final linear ----------------
__global__ void k_zero(float* p, int n) {
  int i = blockIdx.x * blockDim.x + threadIdx.x;
  if (i < n) p[i] = 0.0f;
}

__global__ void k_pool(const float* __restrict__ h, const int* __restrict__ batch,
                       float* pooled, int N) {
  size_t i = (size_t)blockIdx.x * blockDim.x + threadIdx.x;  // float4 index
  if (i < (size_t)N * 32) {
    int n = (int)(i >> 5);
    int fq = (int)(i & 31);
    v4f h4 = ((const v4f*)h)[i];
    float* p = &pooled[(size_t)batch[n] * FDIM + 4 * fq];
    atomicAdd(p + 0, h4.x);
    atomicAdd(p + 1, h4.y);
    atomicAdd(p + 2, h4.z);
    atomicAdd(p + 3, h4.w);
  }
}

__global__ void k_cnt(const int* __restrict__ batch, float* cnt, int N) {
  int n = blockIdx.x * blockDim.x + threadIdx.x;
  if (n < N) atomicAdd(&cnt[batch[n]], 1.0f);
}

__global__ void k_final(const float* __restrict__ pooled, const float* __restrict__ cnt,
                        const float* __restrict__ Wlin, const float* __restrict__ blin,
                        float* __restrict__ out) {
  __shared__ float s0[128];
  __shared__ float s1[128];
  int g = blockIdx.x, f = threadIdx.x;
  float c = cnt[g];
  float p = (c > 0.0f) ? pooled[(size_t)g * FDIM + f] / c : 0.0f;
  s0[f] = p * Wlin[f * 2 + 0];
  s1[f] = p * Wlin[f * 2 + 1];
  __syncthreads();
  for (int s = 64; s > 0; s >>= 1) {
    if (f < s) { s0[f] += s0[f + s]; s1[f] += s1[f + s]; }
    __syncthreads();
  }
  if (f == 0) {
    out[g * 2 + 0] = s0[0] + blin[0];
    out[g * 2 + 1] = s1[0] + blin[1];
  }
}

// ---------------- launch ----------------
extern "C" void kernel_launch(void* const* d_in, const int* in_sizes, int n_in,
                              void* d_out, int out_size, void* d_ws, size_t ws_size,
                              hipStream_t stream) {
  const float* x    = (const float*)d_in[0];
  const int*   ei   = (const int*)d_in[1];  // [2, E]
  const int*   bat  = (const int*)d_in[2];  // [N]
  const float* W1   = (const float*)d_in[3];
  const float* b1   = (const float*)d_in[4];
  const float* W2   = (const float*)d_in[5];
  const float* b2   = (const float*)d_in[6];
  const float* Wlin = (const float*)d_in[7];
  const float* blin = (const float*)d_in[8];
  float* out = (float*)d_out;

  const int N = in_sizes[0] / FDIM;
  const int E = in_sizes[1] / 2;
  const int G = out_size / 2;
  const int* src = ei;
  const int* dst = ei + E;

  float* hA     = (float*)d_ws;               // N*128
  float* hB     = hA + (size_t)N * FDIM;      // N*128
  float* dinv   = hB + (size_t)N * FDIM;      // N (deg -> dinv in place)
  float* pooled = dinv + N;                   // G*128
  float* cnt    = pooled + (size_t)G * FDIM;  // G

  const dim3 blk(256);
  const int gN    = (N + 255) / 256;
  const int gE    = (E + 255) / 256;
  const int gNF4  = (int)(((long long)N * 32 + 255) / 256);
  const int gEF4  = (int)(((long long)E * 32 + 255) / 256);
  const int gGemm = (N + 127) / 128;

  // degree^-1/2 (self-loops included)
  k_deg_init<<<gN, blk, 0, stream>>>(dinv, N);
  k_deg_count<<<gE, blk, 0, stream>>>(dst, dinv, E);
  k_dinv<<<gN, blk, 0, stream>>>(dinv, N);

  // layer 1: h = relu(scatter(x @ W1) + b1)
  k_gemm128<<<gGemm, blk, 0, stream>>>(x, W1, hA, N);
  k_selfloop<<<gNF4, blk, 0, stream>>>(hA, dinv, hB, N);
  k_edge_scatter<<<gEF4, blk, 0, stream>>>(hA, src, dst, dinv, hB, (long long)E);
  k_bias_relu<<<gNF4, blk, 0, stream>>>(hB, b1, hA, N);

  // layer 2
  k_gemm128<<<gGemm, blk, 0, stream>>>(hA, W2, hB, N);
  k_selfloop<<<gNF4, blk, 0, stream>>>(hB, dinv, hA, N);
  k_edge_scatter<<<gEF4, blk, 0, stream>>>(hB, src, dst, dinv, hA, (long long)E);
  k_bias_relu<<<gNF4, blk, 0, stream>>>(hA, b2, hB, N);

  // global mean pool + final linear
  k_zero<<<(G * FDIM + 255) / 256, blk, 0, stream>>>(pooled, G * FDIM);
  k_zero<<<(G + 255) / 256, blk, 0, stream>>>(cnt, G);
  k_pool<<<gNF4, blk, 0, stream>>>(hB, bat, pooled, N);
  k_cnt<<<gN, blk, 0, stream>>>(bat, cnt, N);
  k_final<<<G, dim3(128), 0, stream>>>(pooled, cnt, Wlin, blin, out);
}